// AdaptiveSparseAttention_65996467471045
// MI455X (gfx1250) — compile-verified
//
#include <hip/hip_runtime.h>
#include <hip/hip_bf16.h>

typedef float v2f __attribute__((ext_vector_type(2)));
typedef float v4f __attribute__((ext_vector_type(4)));
typedef float v8f __attribute__((ext_vector_type(8)));
typedef unsigned int v4u __attribute__((ext_vector_type(4)));
typedef int v4i __attribute__((ext_vector_type(4)));
typedef int v8i __attribute__((ext_vector_type(8)));

#define BB 2
#define SS 1536
#define DD 1024
#define HH 16
#define HD 64
#define D3 (3 * DD)
#define KKEEP 1075           // int(S * (1 - 0.3))
#define NEGV (-1e9f)
#define HALF_WIN 16          // WINDOW // 2
#define SCP 1544             // padded score-row pitch (floats)

// ---------------------------------------------------------------------------
// helpers
// ---------------------------------------------------------------------------
__device__ __forceinline__ unsigned sortable(float f) {
  unsigned u = __float_as_uint(f);
  return (u & 0x80000000u) ? ~u : (u | 0x80000000u);
}

// D = A(16x4,f32) * B(4x16,f32) + C(16x16,f32)   -> v_wmma_f32_16x16x4_f32
__device__ __forceinline__ v8f wmma4(v2f a, v2f b, v8f c) {
  return __builtin_amdgcn_wmma_f32_16x16x4_f32(false, a, false, b,
                                               (short)0, c, false, false);
}

// ---------------------------------------------------------------------------
// Tensor Data Mover: 2D f32 tile load Global -> LDS  (CDNA5 ISA)
//   tile_d0 elements along dim0 (contiguous), tile_d1 rows, row stride
//   `stride0` elements.  LDS padding: pad_amount DWORDS-code `padA` every
//   2^(padI) * 8 bytes (padI=4, padA=3 => +4 DWORDs per 32 DWORDs => pitch 36)
//   This toolchain exposes the 6-arg builtin:
//     (v4u g0, v8i g1, v4i g2, v4i g3, v8i g4, i32 cpol)
// ---------------------------------------------------------------------------
__device__ __forceinline__ void tdm_load_2d(const void* gptr, unsigned lds_off,
                                            unsigned tensor_d0,
                                            unsigned tensor_d1,
                                            unsigned stride0, unsigned tile_d0,
                                            unsigned tile_d1, unsigned padI,
                                            unsigned padA) {
  unsigned long long ga = (unsigned long long)gptr;
  v4u g0;
  g0.x = 1u;                                   // count=1 (valid user D#)
  g0.y = lds_off;                              // LDS byte address
  g0.z = (unsigned)(ga & 0xFFFFFFFFull);       // global_addr[31:0]
  g0.w = (unsigned)((ga >> 32) & 0x01FFFFFFull) | (2u << 30);  // addr[56:32]|type=2

  v8i g1;
  g1[0] = (int)((2u << 16)                     // data_size = 4 bytes
                | (1u << 20)                   // pad_enable
                | (padI << 22) | (padA << 25));
  g1[1] = (int)((tensor_d0 & 0xFFFFu) << 16);  // abar_addr=0 | tensor_dim0[15:0]
  g1[2] = (int)((tensor_d0 >> 16) | ((tensor_d1 & 0xFFFFu) << 16));
  g1[3] = (int)((tensor_d1 >> 16) | (tile_d0 << 16));
  g1[4] = (int)(tile_d1 & 0xFFFFu);            // tile_dim1 | tile_dim2=0
  g1[5] = (int)stride0;                        // tensor_dim0_stride[31:0]
  g1[6] = 0;                                   // stride0[47:32]=0 | stride1 lo
  g1[7] = 0;

  v4i z4 = {0, 0, 0, 0};
  v8i z8 = {0, 0, 0, 0, 0, 0, 0, 0};
  __builtin_amdgcn_tensor_load_to_lds(g0, g1, z4, z4, z8, 0);
}

__device__ __forceinline__ unsigned lds_offset_of(const void* p) {
  return (unsigned)(unsigned long long)p;      // low 32 bits = LDS byte offset
}

// ---------------------------------------------------------------------------
// Kernel 0: content-based pattern weights  (B,3) = softmax(MLP(mean_s(x)))
// ---------------------------------------------------------------------------
__global__ __launch_bounds__(256) void pattern_kernel(
    const float* __restrict__ x, const float* __restrict__ w1,
    const float* __restrict__ b1, const float* __restrict__ w2,
    const float* __restrict__ b2, float* __restrict__ pw) {
  __shared__ float pooled[DD];
  __shared__ float hbuf[DD / 4];
  __shared__ float lg[3];
  const int b = blockIdx.x, tid = threadIdx.x;

  for (int d = tid; d < DD; d += 256) {
    float s = 0.f;
    const float* xp = x + (size_t)b * SS * DD + d;
    for (int si = 0; si < SS; ++si) s += xp[(size_t)si * DD];
    pooled[d] = s * (1.f / SS);
  }
  __syncthreads();
  for (int i = tid; i < DD / 4; i += 256) {
    float s = b1[i];
    const float* wr = w1 + (size_t)i * DD;
    for (int d = 0; d < DD; ++d) s += wr[d] * pooled[d];
    hbuf[i] = s > 0.f ? s : 0.f;   // relu
  }
  __syncthreads();
  if (tid < 3) {
    float s = b2[tid];
    const float* wr = w2 + (size_t)tid * (DD / 4);
    for (int i = 0; i < DD / 4; ++i) s += wr[i] * hbuf[i];
    lg[tid] = s;
  }
  __syncthreads();
  if (tid == 0) {
    float m = fmaxf(lg[0], fmaxf(lg[1], lg[2]));
    float e0 = __expf(lg[0] - m), e1 = __expf(lg[1] - m), e2 = __expf(lg[2] - m);
    float inv = 1.f / (e0 + e1 + e2);
    pw[b * 3 + 0] = e0 * inv;
    pw[b * 3 + 1] = e1 * inv;
    pw[b * 3 + 2] = e2 * inv;
  }
}

// ---------------------------------------------------------------------------
// Kernel 1/3: C(MxN) = A(MxK) * Wt(NxK)^T + bias   (row-major, K%32==0)
// block tile 64x128, 256 threads (8 waves), f32 WMMA.
// K=32 slices staged by the Tensor Data Mover (double-buffered):
//   wave0 posts tensor_load_to_lds for slice k+1, all waves run WMMA on
//   slice k, handoff via s_wait_tensorcnt + workgroup barrier.
// ---------------------------------------------------------------------------
__global__ __launch_bounds__(256) void gemm_nt(
    const float* __restrict__ A, const float* __restrict__ Wt,
    const float* __restrict__ bias, float* __restrict__ C,
    int M, int N, int K) {
  __shared__ float Xs[2][64][36];    // 64x32 tile, TDM-padded pitch 36
  __shared__ float Ws[2][128][36];   // 128x32 tile
  const int tid = threadIdx.x;
  const int lane = tid & 31;
  const int wv = tid >> 5;
  const int m0 = blockIdx.x * 64;
  const int n0 = blockIdx.y * 128;
  const int rg = wv >> 1;                // wave row group 0..3  (16 rows each)
  const int cg = wv & 1;                 // wave col group 0..1  (64 cols each)
  const int cs = (lane < 16) ? 0 : 2;    // A/B fragment K-phase per half-wave
  const int l15 = lane & 15;
  const int mo = (lane < 16) ? 0 : 8;    // C/D fragment row offset

  const unsigned ldsX = lds_offset_of(&Xs[0][0][0]);
  const unsigned ldsW = lds_offset_of(&Ws[0][0][0]);
  const unsigned xBytes = 64u * 36u * 4u;
  const unsigned wBytes = 128u * 36u * 4u;

  v8f acc[4];
  for (int j = 0; j < 4; ++j) acc[j] = (v8f){0.f,0.f,0.f,0.f,0.f,0.f,0.f,0.f};

  // prologue: DMA first K-slice into buffer 0
  if (wv == 0) {
    tdm_load_2d(&A[(size_t)m0 * K], ldsX, (unsigned)K, (unsigned)M,
                (unsigned)K, 32u, 64u, 4u, 3u);
    tdm_load_2d(&Wt[(size_t)n0 * K], ldsW, (unsigned)K, (unsigned)N,
                (unsigned)K, 32u, 128u, 4u, 3u);
    __builtin_amdgcn_s_wait_tensorcnt(0);
  }
  __syncthreads();

  int cur = 0;
  for (int kt = 0; kt < K; kt += 32) {
    if (kt + 32 < K && wv == 0) {        // post DMA for next slice -> buf^1
      tdm_load_2d(&A[(size_t)m0 * K + kt + 32], ldsX + (cur ^ 1) * xBytes,
                  (unsigned)K, (unsigned)M, (unsigned)K, 32u, 64u, 4u, 3u);
      tdm_load_2d(&Wt[(size_t)n0 * K + kt + 32], ldsW + (cur ^ 1) * wBytes,
                  (unsigned)K, (unsigned)N, (unsigned)K, 32u, 128u, 4u, 3u);
    }
    const float(*Xc)[36] = Xs[cur];
    const float(*Wc)[36] = Ws[cur];
#pragma unroll
    for (int kk = 0; kk < 32; kk += 4) {
      v2f a = *(const v2f*)&Xc[rg * 16 + l15][kk + cs];
#pragma unroll
      for (int j = 0; j < 4; ++j) {
        v2f bf = *(const v2f*)&Wc[cg * 64 + j * 16 + l15][kk + cs];
        acc[j] = wmma4(a, bf, acc[j]);
      }
    }
    __syncthreads();                     // everyone done reading buf[cur]
    if (kt + 32 < K) {
      if (wv == 0) __builtin_amdgcn_s_wait_tensorcnt(0);   // next slice landed
      __syncthreads();                   // publish to all waves
      cur ^= 1;
    }
  }
#pragma unroll
  for (int j = 0; j < 4; ++j) {
    int n = n0 + cg * 64 + j * 16 + l15;
    float bv = bias ? bias[n] : 0.f;
#pragma unroll
    for (int vi = 0; vi < 8; ++vi) {
      int m = m0 + rg * 16 + vi + mo;
      C[(size_t)m * N + n] = acc[j][vi] + bv;
    }
  }
}

// ---------------------------------------------------------------------------
// Kernel 2: fused attention for one (b, h, 16-row q tile)
//   phase 1: scores 16x1536 = (Q K^T) * scale  via WMMA -> LDS
//   phase 2: exact kth-largest (k=1075) per row of (score*sw+sb) via
//            sortable-uint binary search, wave32 shuffle reductions
//   phase 3: local-band + learned-sparse mask, softmax numerators in LDS
//   phase 4: O = P V via WMMA (keys split over 8 waves) + LDS reduction
// ---------------------------------------------------------------------------
__global__ __launch_bounds__(256) void attn_kernel(
    const float* __restrict__ qkv, const float* __restrict__ pw,
    const float* __restrict__ sparse_w, const float* __restrict__ sparse_b,
    float* __restrict__ att) {
  extern __shared__ char smem[];
  float* qs = (float*)smem;                  // 16 x 68
  float* sc = qs + 16 * 68;                  // 16 x SCP score/prob tile
  float* op = sc + 16 * SCP;                 // 8 x 16 x 64 wave partials
  unsigned* thr = (unsigned*)(op + 8 * 16 * 64);  // 16 row thresholds
  float* rsum = (float*)(thr + 16);          // 16 row softmax denominators

  const int tid = threadIdx.x, lane = tid & 31, wv = tid >> 5;
  const int bx = blockIdx.x;
  const int b = bx / (HH * (SS / 16));
  const int rem = bx % (HH * (SS / 16));
  const int h = rem / (SS / 16);
  const int q0 = (rem % (SS / 16)) * 16;

  const float scale = 0.125f;                // HD^-0.5
  const float wl = pw[b * 3 + 0];
  const float wsp = pw[b * 3 + 2];
  const float sw = sparse_w[h];
  const float sb = sparse_b[h];
  const float* base = qkv + (size_t)b * SS * D3;

  const int cs = (lane < 16) ? 0 : 2;
  const int l15 = lane & 15;
  const int mo = (lane < 16) ? 0 : 8;

  // ---- stage Q tile (16x64) --------------------------------------------
  {
    int idx = tid * 4;                       // 1024 floats total
    int m = idx >> 6, d = idx & 63;
    *(v4f*)&qs[m * 68 + d] =
        *(const v4f*)&base[(size_t)(q0 + m) * D3 + h * HD + d];
  }
  __syncthreads();

  // ---- phase 1: scores -------------------------------------------------
  for (int cb = 0; cb < 12; ++cb) {          // wave covers 192 key columns
    const int col0 = wv * 192 + cb * 16;
    v8f acc = (v8f){0.f,0.f,0.f,0.f,0.f,0.f,0.f,0.f};
    const float* kb = base + (size_t)(col0 + l15) * D3 + DD + h * HD;
#pragma unroll
    for (int kk = 0; kk < HD; kk += 4) {
      v2f a = *(const v2f*)&qs[l15 * 68 + kk + cs];
      v2f bf = *(const v2f*)&kb[kk + cs];    // B[k,n] = K[n,k] (K^T)
      acc = wmma4(a, bf, acc);
    }
#pragma unroll
    for (int vi = 0; vi < 8; ++vi)
      sc[(vi + mo) * SCP + col0 + l15] = acc[vi] * scale;
  }
  __syncthreads();

  // ---- phase 2: kth-largest threshold per row (2 rows / wave) ----------
  for (int rr = 0; rr < 2; ++rr) {
    const int row = wv * 2 + rr;
    unsigned key[48];
#pragma unroll
    for (int j = 0; j < 48; ++j) {
      float sp = sc[row * SCP + lane + j * 32] * sw + sb;
      key[j] = sortable(sp);
    }
    unsigned lo = 0u, hi = 0xFFFFFFFFu;      // max u with count(>=u) >= KKEEP
    while (lo < hi) {
      unsigned span = hi - lo;
      unsigned mid = lo + (span >> 1) + (span & 1u);   // in (lo, hi]
      int c = 0;
#pragma unroll
      for (int j = 0; j < 48; ++j) c += (key[j] >= mid) ? 1 : 0;
      for (int mm = 16; mm >= 1; mm >>= 1) c += __shfl_xor(c, mm, 32);
      if (c >= KKEEP) lo = mid; else hi = mid - 1u;
    }
    if (lane == 0) thr[row] = lo;
  }
  __syncthreads();

  // ---- phase 3: masks + softmax numerators -----------------------------
  for (int rr = 0; rr < 2; ++rr) {
    const int row = wv * 2 + rr;
    const unsigned t = thr[row];
    const int qg = q0 + row;
    float lv[48];
    float mx = -3.4e38f;
#pragma unroll
    for (int j = 0; j < 48; ++j) {
      int col = lane + j * 32;
      float s = sc[row * SCP + col];
      float add = 0.f;
      int dlt = col - qg; if (dlt < 0) dlt = -dlt;
      if (dlt > HALF_WIN) add += wl * NEGV;              // local band mask
      if (sortable(s * sw + sb) < t) add += wsp * NEGV;  // learned sparse mask
      float l = s + add;                                  // TEMP == 1
      lv[j] = l;
      mx = fmaxf(mx, l);
    }
    for (int mm = 16; mm >= 1; mm >>= 1) mx = fmaxf(mx, __shfl_xor(mx, mm, 32));
    float sum = 0.f;
#pragma unroll
    for (int j = 0; j < 48; ++j) {
      float e = __expf(lv[j] - mx);
      sc[row * SCP + lane + j * 32] = e;
      sum += e;
    }
    for (int mm = 16; mm >= 1; mm >>= 1) sum += __shfl_xor(sum, mm, 32);
    if (lane == 0) rsum[row] = sum;
  }
  __syncthreads();

  // ---- phase 4: O = P V  (wave handles 192 keys) -----------------------
  {
    v8f acc[4];
    for (int j = 0; j < 4; ++j) acc[j] = (v8f){0.f,0.f,0.f,0.f,0.f,0.f,0.f,0.f};
    const int kbeg = wv * 192;
    const float* vbase = base + 2 * DD + h * HD;
    for (int kk = 0; kk < 192; kk += 4) {
      v2f a = *(const v2f*)&sc[l15 * SCP + kbeg + kk + cs];
#pragma unroll
      for (int j = 0; j < 4; ++j) {
        int d = j * 16 + l15;
        v2f bf;                              // B[k,n] = V[key, d]
        bf.x = vbase[(size_t)(kbeg + kk + cs) * D3 + d];
        bf.y = vbase[(size_t)(kbeg + kk + cs + 1) * D3 + d];
        acc[j] = wmma4(a, bf, acc[j]);
      }
    }
#pragma unroll
    for (int j = 0; j < 4; ++j)
#pragma unroll
      for (int vi = 0; vi < 8; ++vi)
        op[wv * 1024 + (vi + mo) * 64 + j * 16 + l15] = acc[j][vi];
  }
  __syncthreads();

  // ---- cross-wave reduction + normalize + store ------------------------
  for (int i = 0; i < 4; ++i) {
    int idx = tid + i * 256;                 // 16*64 = 1024 elements
    int m = idx >> 6, d = idx & 63;
    float s = 0.f;
#pragma unroll
    for (int w = 0; w < 8; ++w) s += op[w * 1024 + idx];
    att[(size_t)(b * SS + q0 + m) * DD + h * HD + d] = s / rsum[m];
  }
}

// ---------------------------------------------------------------------------
// host launcher
// ---------------------------------------------------------------------------
extern "C" void kernel_launch(void* const* d_in, const int* in_sizes, int n_in,
                              void* d_out, int out_size, void* d_ws,
                              size_t ws_size, hipStream_t stream) {
  const float* x        = (const float*)d_in[0];
  const float* w_qkv    = (const float*)d_in[1];
  const float* w_proj   = (const float*)d_in[2];
  const float* b_proj   = (const float*)d_in[3];
  const float* w_ps1    = (const float*)d_in[4];
  const float* b_ps1    = (const float*)d_in[5];
  const float* w_ps2    = (const float*)d_in[6];
  const float* b_ps2    = (const float*)d_in[7];
  const float* sparse_w = (const float*)d_in[8];
  const float* sparse_b = (const float*)d_in[9];
  float* out = (float*)d_out;

  // workspace layout (bytes): qkv | att | pattern-weights
  float* qkv = (float*)d_ws;                              // 2*1536*3072 f32
  float* att = qkv + (size_t)BB * SS * D3;                // 2*1536*1024 f32
  float* pw  = att + (size_t)BB * SS * DD;                // 2*3 f32

  // 0) pattern weights
  pattern_kernel<<<BB, 256, 0, stream>>>(x, w_ps1, b_ps1, w_ps2, b_ps2, pw);

  // 1) qkv = x @ w_qkv^T      (3072x1024) x (3072x1024)^T -> 3072x3072
  {
    dim3 g((BB * SS) / 64, D3 / 128);
    gemm_nt<<<g, 256, 0, stream>>>(x, w_qkv, nullptr, qkv,
                                   BB * SS, D3, DD);
  }

  // 2) fused attention
  {
    const size_t smem = (size_t)(16 * 68 + 16 * SCP + 8 * 16 * 64) * 4 +
                        16 * 4 /*thr*/ + 16 * 4 /*rsum*/;
    dim3 g(BB * HH * (SS / 16));
    attn_kernel<<<g, 256, smem, stream>>>(qkv, pw, sparse_w, sparse_b, att);
  }

  // 3) out = att @ w_proj^T + b_proj   (3072x1024) x (1024x1024)^T
  {
    dim3 g((BB * SS) / 64, DD / 128);
    gemm_nt<<<g, 256, 0, stream>>>(att, w_proj, b_proj, out,
                                   BB * SS, DD, DD);
  }
}